// NADE_9070970929769
// MI455X (gfx1250) — compile-verified
//
#include <hip/hip_runtime.h>
#include <hip/hip_bf16.h>
#include <math.h>

#define B_ 16384
#define D_ 784
#define H_ 1024

typedef __attribute__((ext_vector_type(2))) float v2f;
typedef __attribute__((ext_vector_type(8))) float v8f;

// sigmoid via hardware v_exp_f32 + v_rcp_f32 (~1 ulp each; well within tolerance)
__device__ __forceinline__ float sigmoidf(float z) {
    return __builtin_amdgcn_rcpf(1.0f + __expf(-z));
}

// -------------------------------------------------------------------------
// Precompute kernel: one wave (32 threads) per 16 consecutive output dims d.
//   S0[d] = dot(V[d,:], s)        + b[d]*sum(s),       s[h]   = sigmoid(c[h])
//   S1[d] = dot(V[d,:], t_{d-1})  + b[d]*sum(t_{d-1}), t_e[h] = sigmoid(W[h,e]+c[h])
// The 16 dot products are the diagonal of a 16x16 WMMA tile:
//   A row m      = V[d0+m, :]                (16x4 f32 fragments over K=H)
//   B col n (S1) = t_{d0+n-1}                (4x16 f32 fragments)
//   B col n (S0) = s  (same for every n)
// -------------------------------------------------------------------------
__global__ __launch_bounds__(32) void nade_precompute(
    const float* __restrict__ V, const float* __restrict__ b,
    const float* __restrict__ W, const float* __restrict__ c,
    float* __restrict__ S0, float* __restrict__ S1)
{
    const int lane = threadIdx.x;       // 0..31
    const int half = lane >> 4;         // 0: K=0,1   1: K=2,3
    const int l16  = lane & 15;         // A row m / B col n
    const int d0   = blockIdx.x * 16;

    const float* Vrow = V + (size_t)(d0 + l16) * H_;   // A-matrix row
    int dcol = d0 + l16 - 1;                           // t-column for B (S1)
    if (dcol < 0) dcol = 0;                            // S1[0] is never consumed

    v8f acc0 = {0.f,0.f,0.f,0.f,0.f,0.f,0.f,0.f};
    v8f acc1 = {0.f,0.f,0.f,0.f,0.f,0.f,0.f,0.f};
    float colsum = 0.0f;   // this lane's K-share of sum_h t_{dcol}[h]
    float csum   = 0.0f;   // this lane's K-share of sum_h s[h]

    for (int k = 0; k < H_ / 4; ++k) {
        const int h = 4 * k + 2 * half;
        // A fragment: lanes 0-15 hold K={0,1}, lanes 16-31 hold K={2,3}
        v2f a; a.x = Vrow[h]; a.y = Vrow[h + 1];
        const float c0 = c[h], c1 = c[h + 1];
        // B fragment (S1 path): column n = l16 sources t_{d0+n-1}
        v2f b1;
        b1.x = sigmoidf(W[(size_t)h       * D_ + dcol] + c0);
        b1.y = sigmoidf(W[(size_t)(h + 1) * D_ + dcol] + c1);
        // B fragment (S0 path): every column = s = sigmoid(c)
        v2f b0;
        b0.x = sigmoidf(c0);
        b0.y = sigmoidf(c1);
        colsum += b1.x + b1.y;
        csum   += b0.x + b0.y;
        acc0 = __builtin_amdgcn_wmma_f32_16x16x4_f32(false, a, false, b0,
                                                     (short)0, acc0, false, false);
        acc1 = __builtin_amdgcn_wmma_f32_16x16x4_f32(false, a, false, b1,
                                                     (short)0, acc1, false, false);
    }

    // Diagonal extraction per C/D layout:
    //   m = 0..7  -> lane m      , VGPR m
    //   m = 8..15 -> lane m + 16 , VGPR m - 8
    __shared__ float sd0[16], sd1[16], scol[32], scs[32];
    scol[lane] = colsum;
    scs[lane]  = csum;
    const int idx = (lane < 8) ? lane : ((lane >= 24) ? (lane - 24) : -1);
    float dv0 = 0.0f, dv1 = 0.0f;
    #pragma unroll
    for (int r = 0; r < 8; ++r) {
        if (r == idx) { dv0 = acc0[r]; dv1 = acc1[r]; }
    }
    if (idx >= 0) {
        const int m = (lane < 8) ? lane : (lane - 16);
        sd0[m] = dv0;
        sd1[m] = dv1;
    }
    __syncthreads();

    if (lane < 16) {
        const int d   = d0 + lane;
        const float bd = b[d];
        const float tsum = scol[lane] + scol[lane + 16];  // sum_h t_{d-1}[h]
        const float Ssum = scs[0] + scs[16];              // sum_h s[h]
        S0[d] = sd0[lane] + bd * Ssum;
        S1[d] = sd1[lane] + bd * tsum;
    }
}

// -------------------------------------------------------------------------
// Main pass: HBM-bandwidth-bound elementwise map (~102 MB total traffic).
//   out[b,d] = sigmoid( x[b,d-1] ? S1[d] : S0[d] ),  out[b,0] = sigmoid(S0[0])
// Each thread handles one d for 4 consecutive batch rows: S0/S1 loaded once,
// x/out traffic fully coalesced along d within each row.
// -------------------------------------------------------------------------
__global__ __launch_bounds__(256) void nade_map(
    const float* __restrict__ x, const float* __restrict__ S0,
    const float* __restrict__ S1, float* __restrict__ out)
{
    const int d  = blockIdx.x * blockDim.x + threadIdx.x;
    const int b0 = blockIdx.y * 4;
    if (d >= D_) return;

    const float s0 = S0[d];
    const float s1 = (d != 0) ? S1[d] : s0;
    const size_t base = (size_t)b0 * D_ + d;

    #pragma unroll
    for (int i = 0; i < 4; ++i) {
        const size_t o = base + (size_t)i * D_;
        float xv = 0.0f;
        if (d != 0) xv = x[o - 1];                 // exactly 0.0f or 1.0f
        const float pre = (xv > 0.5f) ? s1 : s0;
        out[o] = sigmoidf(pre);
    }
}

extern "C" void kernel_launch(void* const* d_in, const int* in_sizes, int n_in,
                              void* d_out, int out_size, void* d_ws, size_t ws_size,
                              hipStream_t stream)
{
    const float* x = (const float*)d_in[0];   // (B, D)
    const float* V = (const float*)d_in[1];   // (D, H)
    const float* b = (const float*)d_in[2];   // (D,)
    const float* W = (const float*)d_in[3];   // (H, D)
    const float* c = (const float*)d_in[4];   // (1, H)
    float* out = (float*)d_out;               // (B, D)

    float* S0 = (float*)d_ws;                 // D floats
    float* S1 = S0 + D_;                      // D floats

    nade_precompute<<<dim3(D_ / 16), dim3(32), 0, stream>>>(V, b, W, c, S0, S1);
    nade_map<<<dim3((D_ + 255) / 256, B_ / 4), dim3(256), 0, stream>>>(x, S0, S1, out);
}